// YOLOv1Loss2D_36524401885474
// MI455X (gfx1250) — compile-verified
//
#include <hip/hip_runtime.h>

// YOLOv1 loss, MI455X (gfx1250).
// Roofline: 2 x 16384 x 49 x 30 x 4 B = 192.7 MB streamed once -> ~8.3 us at
// 23.3 TB/s; ~0.12 GFLOP of VALU -> pure bandwidth problem.
// Structure: persistent blocks, double-buffered TDM (tensor_load_to_lds) tile
// pipeline Global->LDS, one cell per lane out of LDS (bank-conflict-free),
// exact f32 WMMA 16x16x4 wave reduction, deterministic two-pass finish.

#define CH           30
#define TILE_CELLS   128
#define TILE_ELEMS   (TILE_CELLS * CH)   // 3840 f32 = 15360 B per tensor tile
#define TOTAL_TILES  6272                // 16384*49 / 128
#define NBLOCKS      784                 // 6272 / 784 = exactly 8 tiles/block
#define TILES_PER_BLOCK 8
#define LAMBDA_COORD 5.0f
#define LAMBDA_NOOBJ 0.5f
#define EPS_WH       1e-6f
#define EPS_IOU      1e-6f
#define INV_BATCH    (1.0f / 16384.0f)

typedef unsigned int u32x4 __attribute__((ext_vector_type(4)));
typedef int          i32x8 __attribute__((ext_vector_type(8)));
typedef int          i32x4 __attribute__((ext_vector_type(4)));
typedef float        v2f   __attribute__((ext_vector_type(2)));
typedef float        v8f   __attribute__((ext_vector_type(8)));

// ---- TDM: 1-D contiguous f32 copy Global -> LDS (D# per CDNA5 ISA)
// amdgpu-toolchain builtin (6 args): (u32x4, i32x8, i32x4, i32x4, i32x8, i32 cpol)
__device__ __forceinline__ void tdm_load_1d_f32(const float* gptr,
                                                unsigned lds_byte,
                                                unsigned nelem) {
  unsigned long long ga = (unsigned long long)(uintptr_t)gptr;
  u32x4 g0;
  g0.x = 1u;                                          // count=1, user descriptor
  g0.y = lds_byte;                                    // lds_addr [63:32]
  g0.z = (unsigned)ga;                                // global_addr[31:0]
  g0.w = (unsigned)((ga >> 32) & 0x01FFFFFFull)       // global_addr[56:32]
       | (2u << 30);                                  // type=2 ("image")
  i32x8 g1;
  g1[0] = (int)(2u << 16);                            // data_size=2 (4 B elems)
  g1[1] = (int)((nelem & 0xFFFFu) << 16);             // tensor_dim0[15:0]
  g1[2] = (int)((nelem >> 16) & 0xFFFFu);             // tensor_dim0[31:16], dim1=0
  g1[3] = (int)((nelem & 0xFFFFu) << 16);             // tile_dim0
  g1[4] = 0;                                          // tile_dim1/2 unused
  g1[5] = (int)nelem;                                 // tensor_dim0_stride
  g1[6] = 0;
  g1[7] = 0;
  i32x4 z4 = {0, 0, 0, 0};
  i32x8 z8 = {0, 0, 0, 0, 0, 0, 0, 0};
  __builtin_amdgcn_tensor_load_to_lds(g0, g1, z4, z4, z8, 0);
}

// ---- Exact 32-lane f32 sum, one WMMA + lane-local adds + one swizzle.
// A holds partials (a = {x,0}): A[m][0]=p[m], A[m][2]=p[m+16]; B = ones.
// D[m][n] = p[m] + p[m+16]. Summing the 8 C VGPRs lane-locally gives
// lanes 0-15:  sum(p[0..7])  + sum(p[16..23])
// lanes 16-31: sum(p[8..15]) + sum(p[24..31])
// -> one SWAPX16 ds_swizzle (0x401F) + add = full wave sum in every lane.
__device__ __forceinline__ float wave_sum32(float x) {
  v2f a; a[0] = x;    a[1] = 0.0f;
  v2f b; b[0] = 1.0f; b[1] = 1.0f;
  v8f c = {};
  c = __builtin_amdgcn_wmma_f32_16x16x4_f32(false, a, false, b,
                                            (short)0, c, false, false);
  float s = ((c[0] + c[1]) + (c[2] + c[3])) + ((c[4] + c[5]) + (c[6] + c[7]));
  int sw = __builtin_amdgcn_ds_swizzle(__float_as_int(s), 0x401F); // SWAPX16
  return s + __int_as_float(sw);
}

__device__ __forceinline__ float iou_yolo(float ax, float ay, float aw, float ah,
                                          float bx, float by, float bw, float bh) {
  float ax1 = ax - aw * 0.5f, ax2 = ax + aw * 0.5f;
  float ay1 = ay - ah * 0.5f, ay2 = ay + ah * 0.5f;
  float bx1 = bx - bw * 0.5f, bx2 = bx + bw * 0.5f;
  float by1 = by - bh * 0.5f, by2 = by + bh * 0.5f;
  float iw = fmaxf(fminf(ax2, bx2) - fmaxf(ax1, bx1), 0.0f);
  float ih = fmaxf(fminf(ay2, by2) - fmaxf(ay1, by1), 0.0f);
  float inter = iw * ih;
  float a1 = fabsf((ax2 - ax1) * (ay2 - ay1));
  float a2 = fabsf((bx2 - bx1) * (by2 - by1));
  return inter / (a1 + a2 - inter + EPS_IOU);
}

__device__ __forceinline__ float cell_loss(const float* __restrict__ yt,
                                           const float* __restrict__ yp) {
  const float bx = yt[0], by = yt[1], bw = yt[2], bh = yt[3];
  const float yt4 = yt[4];
  const float objf = (yt4 == 1.0f) ? 1.0f : 0.0f;
  const float noobjf = 1.0f - objf;

  const float p0 = yp[0], p1 = yp[1], p2 = yp[2], p3 = yp[3], c1 = yp[4];
  const float q0 = yp[5], q1 = yp[6], q2 = yp[7], q3 = yp[8], c2 = yp[9];

  const float iou1 = iou_yolo(bx, by, bw, bh, p0, p1, p2, p3);
  const float iou2 = iou_yolo(bx, by, bw, bh, q0, q1, q2, q3);
  const bool best1 = iou1 > iou2;

  const float hx = best1 ? p0 : q0;
  const float hy = best1 ? p1 : q1;
  const float hw = best1 ? p2 : q2;
  const float hh = best1 ? p3 : q3;
  const float hc = best1 ? c1 : c2;

  float dx = bx - hx, dy = by - hy;
  float loss = LAMBDA_COORD * objf * (dx * dx + dy * dy);

  float wt = sqrtf(bw) - sqrtf(fabsf(hw + EPS_WH));
  float ht = sqrtf(bh) - sqrtf(fabsf(hh + EPS_WH));
  loss += LAMBDA_COORD * objf * (wt * wt + ht * ht);

  float dc = yt4 - hc;
  loss += objf * dc * dc;

  // reference quirk: yp[..., 4::5] -> j in {4,9,14,19,24,29}
  float no = 0.0f;
#pragma unroll
  for (int j = 4; j < CH; j += 5) {
    float d = yt4 - yp[j];
    no += d * d;
  }
  loss += LAMBDA_NOOBJ * noobjf * no;

  float cls = 0.0f;
#pragma unroll
  for (int k = 10; k < CH; ++k) {
    float d = yt[k] - yp[k];
    cls += d * d;
  }
  loss += objf * cls;
  return loss;
}

// Persistent double-buffered TDM pipeline: each block walks 8 tiles with a
// grid-stride of NBLOCKS, issuing tile t+step into buf^1 while computing tile
// t from buf. s_wait_tensorcnt(2) retires exactly the oldest DMA pair.
__global__ void __launch_bounds__(TILE_CELLS)
yolo_loss_pipeline(const float* __restrict__ yt_g,
                   const float* __restrict__ yp_g,
                   float* __restrict__ partials) {
  __shared__ float s_buf[2][2][TILE_ELEMS];   // [buffer][tensor][elem] = 60 KB
  __shared__ float s_red[TILE_CELLS / 32];

  const unsigned tid = threadIdx.x;
  unsigned t = blockIdx.x;
  unsigned buf = 0;

  if (tid < 32u) {
    tdm_load_1d_f32(yt_g + (size_t)t * TILE_ELEMS,
                    (unsigned)(uintptr_t)(&s_buf[0][0][0]), TILE_ELEMS);
    tdm_load_1d_f32(yp_g + (size_t)t * TILE_ELEMS,
                    (unsigned)(uintptr_t)(&s_buf[0][1][0]), TILE_ELEMS);
  }

  float accum = 0.0f;
#pragma unroll 1
  for (; t < TOTAL_TILES; t += NBLOCKS) {
    const unsigned nxt = t + NBLOCKS;            // block-uniform
    if (tid < 32u) {
      if (nxt < TOTAL_TILES) {
        tdm_load_1d_f32(yt_g + (size_t)nxt * TILE_ELEMS,
                        (unsigned)(uintptr_t)(&s_buf[buf ^ 1u][0][0]), TILE_ELEMS);
        tdm_load_1d_f32(yp_g + (size_t)nxt * TILE_ELEMS,
                        (unsigned)(uintptr_t)(&s_buf[buf ^ 1u][1][0]), TILE_ELEMS);
        __builtin_amdgcn_s_wait_tensorcnt(2);    // oldest pair (tile t) done
      } else {
        __builtin_amdgcn_s_wait_tensorcnt(0);    // pipeline tail
      }
    }
    __syncthreads();                             // tile t visible to all waves

    // one cell per lane; stride 30 dwords -> conflict-free within wave32
    accum += cell_loss(&s_buf[buf][0][tid * CH], &s_buf[buf][1][tid * CH]);

    __syncthreads();                             // all reads of buf done before
    buf ^= 1u;                                   // it is re-targeted by DMA
  }

  // Block reduction once per block (8 tiles amortized).
  float s = wave_sum32(accum);
  if ((tid & 31u) == 0u) s_red[tid >> 5] = s;
  __syncthreads();
  if (tid == 0u) {
    float tot = 0.0f;
#pragma unroll
    for (int w = 0; w < TILE_CELLS / 32; ++w) tot += s_red[w];
    partials[blockIdx.x] = tot;
  }
}

__global__ void __launch_bounds__(256)
yolo_loss_finish(const float* __restrict__ partials, float* __restrict__ out) {
  __shared__ float s_red[8];
  float acc = 0.0f;
  for (unsigned i = threadIdx.x; i < NBLOCKS; i += 256u) acc += partials[i];
  float s = wave_sum32(acc);
  if ((threadIdx.x & 31u) == 0u) s_red[threadIdx.x >> 5] = s;
  __syncthreads();
  if (threadIdx.x == 0u) {
    float t = 0.0f;
#pragma unroll
    for (int w = 0; w < 8; ++w) t += s_red[w];
    out[0] = t * INV_BATCH;
  }
}

extern "C" void kernel_launch(void* const* d_in, const int* in_sizes, int n_in,
                              void* d_out, int out_size, void* d_ws, size_t ws_size,
                              hipStream_t stream) {
  const float* y_trues = (const float*)d_in[0];
  const float* y_preds = (const float*)d_in[1];
  float* out      = (float*)d_out;
  float* partials = (float*)d_ws;   // NBLOCKS floats = 3.1 KB scratch

  yolo_loss_pipeline<<<NBLOCKS, TILE_CELLS, 0, stream>>>(y_trues, y_preds, partials);
  yolo_loss_finish<<<1, 256, 0, stream>>>(partials, out);
}